// SpatialGraphReprogramming_37580963840353
// MI455X (gfx1250) — compile-verified
//
#include <hip/hip_runtime.h>
#include <math.h>

typedef float v2f __attribute__((ext_vector_type(2)));
typedef float v8f __attribute__((ext_vector_type(8)));

#define NSEQ   325
#define NROWS  166400          /* 512 * 325 */
#define KSTR   66              /* LDS row stride for K/V (bank-conflict pad) */
#define KROWS  336             /* 325 rounded up to 16, zero padded */

/* d_ws float offsets */
#define WEFF_OFF 0             /* 3*4096 effective qkv weights [out][c][j] */
#define BEFF_OFF 12288         /* 192 effective qkv biases */
#define W2_OFF   12480         /* 4096 fused out-proj weight [h][c] */
#define B2_OFF   16576         /* 64 fused out-proj bias */
#define MASK_OFF 16640         /* 325*11 packed mask words */
#define QBUF_OFF 20480
#define ROWBUF   10649600      /* 166400*64 */
#define KBUF_OFF (QBUF_OFF + ROWBUF)
#define VBUF_OFF (QBUF_OFF + 2*ROWBUF)

static __device__ __forceinline__ v8f wmma4(v2f a, v2f b, v8f c) {
  /* D(16x16,f32) += A(16x4,f32) * B(4x16,f32) */
  return __builtin_amdgcn_wmma_f32_16x16x4_f32(false, a, false, b, (short)0, c,
                                               false, false);
}

/* ------------------------------------------------------------------ */
/* Kernel 1: fold projections into 64x64 effective weights             */
/* ------------------------------------------------------------------ */
__global__ void prep_weights(const float* __restrict__ W_in,
                             const float* __restrict__ b_in,
                             const float* __restrict__ ipw,
                             const float* __restrict__ ipb,
                             const float* __restrict__ out_w,
                             const float* __restrict__ out_b,
                             const float* __restrict__ W_out,
                             const float* __restrict__ b_out,
                             float* __restrict__ ws) {
  int tid = threadIdx.x;
  /* Weff[o][c][j] = sum_t W_in[c][t] * ipw[o*64+j][t] */
  for (int idx = tid; idx < 3 * 4096; idx += 256) {
    int o = idx >> 12, rem = idx & 4095, c = rem >> 6, j = rem & 63;
    const float* wrow = ipw + (size_t)(o * 64 + j) * 64;
    float s = 0.f;
    for (int t = 0; t < 64; ++t) s += W_in[c * 64 + t] * wrow[t];
    ws[WEFF_OFF + idx] = s;
  }
  /* beff[o*64+j] = b_in . ipw_row + ipb */
  for (int idx = tid; idx < 192; idx += 256) {
    const float* wrow = ipw + (size_t)idx * 64;
    float s = ipb[idx];
    for (int t = 0; t < 64; ++t) s += b_in[t] * wrow[t];
    ws[BEFF_OFF + idx] = s;
  }
  /* W2[h][c] = sum_j out_w[j][h] * W_out[c][j] */
  for (int idx = tid; idx < 4096; idx += 256) {
    int h = idx >> 6, c = idx & 63;
    float s = 0.f;
    for (int j = 0; j < 64; ++j) s += out_w[j * 64 + h] * W_out[c * 64 + j];
    ws[W2_OFF + idx] = s;
  }
  for (int idx = tid; idx < 64; idx += 256) {
    float s = b_out[idx];
    for (int j = 0; j < 64; ++j) s += out_b[j] * W_out[idx * 64 + j];
    ws[B2_OFF + idx] = s;
  }
}

/* ------------------------------------------------------------------ */
/* Kernel 2: pack adjacency + forced self loops into bitmask           */
/* ------------------------------------------------------------------ */
__global__ void prep_mask(const int* __restrict__ adj, unsigned* __restrict__ m) {
  int idx = blockIdx.x * 256 + threadIdx.x;
  if (idx >= NSEQ * 11) return;
  int nq = idx / 11, w = idx % 11;
  unsigned bits = 0;
  for (int b = 0; b < 32; ++b) {
    int nk = w * 32 + b;
    if (nk < NSEQ && (adj[nq * NSEQ + nk] != 0 || nk == nq)) bits |= (1u << b);
  }
  m[idx] = bits;
}

/* ------------------------------------------------------------------ */
/* Kernel 3: q,k,v = x @ Weff + beff   (166400 x 64 rows, WMMA)        */
/* ------------------------------------------------------------------ */
__global__ void qkv_kernel(const float* __restrict__ x, const float* __restrict__ ws,
                           float* __restrict__ q, float* __restrict__ k,
                           float* __restrict__ v) {
  __shared__ float sW[12480];
  for (int i = threadIdx.x; i < 12480; i += 256) sW[i] = ws[i];
  __syncthreads();
  int wave = threadIdx.x >> 5, lane = threadIdx.x & 31;
  int m16 = lane & 15, koff = (lane >> 4) * 2, rbase = (lane >> 4) * 8;
  long row0 = ((long)blockIdx.x * 8 + wave) * 16;   /* 1300 blocks * 8 waves = 10400 tiles */
  v2f a[16];
  const float* xr = x + (row0 + m16) * 64;
#pragma unroll
  for (int kc = 0; kc < 16; ++kc) a[kc] = *(const v2f*)(xr + kc * 4 + koff);
#pragma unroll
  for (int o = 0; o < 3; ++o) {
    float* dst = (o == 0) ? q : (o == 1) ? k : v;
    const float* W = &sW[o * 4096];
    const float* bias = &sW[12288 + o * 64];
#pragma unroll
    for (int nt = 0; nt < 4; ++nt) {
      int n = nt * 16 + m16;
      float bv = bias[n];
      v8f c = {bv, bv, bv, bv, bv, bv, bv, bv};
#pragma unroll
      for (int kc = 0; kc < 16; ++kc) {
        v2f b = {W[(kc * 4 + koff) * 64 + n], W[(kc * 4 + koff + 1) * 64 + n]};
        c = wmma4(a[kc], b, c);
      }
#pragma unroll
      for (int r = 0; r < 8; ++r) dst[(row0 + rbase + r) * 64 + n] = c[r];
    }
  }
}

/* ------------------------------------------------------------------ */
/* Kernel 4: masked flash attention, one block per bp, K/V in LDS      */
/* ------------------------------------------------------------------ */
__global__ void attn_kernel(const float* __restrict__ qb, const float* __restrict__ kb,
                            const float* __restrict__ vb,
                            const unsigned* __restrict__ maskg,
                            float* __restrict__ ob) {
  extern __shared__ float lds[];
  float* kS = lds;                                    /* 336*66 */
  float* vS = lds + KROWS * KSTR;                     /* 336*66 */
  unsigned* mS = (unsigned*)(lds + 2 * KROWS * KSTR); /* 325*12 */
  float* pS = lds + 2 * KROWS * KSTR + NSEQ * 12;     /* 8 waves * 272 */

  int bp = blockIdx.x;
  const float* kg = kb + (size_t)bp * NSEQ * 64;
  const float* vg = vb + (size_t)bp * NSEQ * 64;
  for (int i = threadIdx.x; i < KROWS * 32; i += 256) {
    int n = i >> 5, d2 = (i & 31) * 2;
    float a0 = 0.f, a1 = 0.f, b0 = 0.f, b1 = 0.f;
    if (n < NSEQ) {
      a0 = kg[n * 64 + d2]; a1 = kg[n * 64 + d2 + 1];
      b0 = vg[n * 64 + d2]; b1 = vg[n * 64 + d2 + 1];
    }
    kS[n * KSTR + d2] = a0; kS[n * KSTR + d2 + 1] = a1;
    vS[n * KSTR + d2] = b0; vS[n * KSTR + d2 + 1] = b1;
  }
  for (int i = threadIdx.x; i < NSEQ * 11; i += 256)
    mS[(i / 11) * 12 + (i % 11)] = maskg[i];
  __syncthreads();

  int wave = threadIdx.x >> 5, lane = threadIdx.x & 31;
  int m16 = lane & 15, half = lane >> 4, koff = half * 2, rbase = half * 8;
  float* pw = pS + wave * 272;                        /* 16x16 P tile, stride 17 */

  for (int qt = wave; qt < 21; qt += 8) {
    int q0 = qt * 16;
    v2f qa[4][4];
    {
      int qn = q0 + m16; if (qn > NSEQ - 1) qn = NSEQ - 1;
      const float* qr = qb + ((size_t)bp * NSEQ + qn) * 64;
#pragma unroll
      for (int h = 0; h < 4; ++h)
#pragma unroll
        for (int kc = 0; kc < 4; ++kc)
          qa[h][kc] = *(const v2f*)(qr + h * 16 + kc * 4 + koff);
    }
    float mrow[4][8], lrow[4][8];
    v8f oacc[4];
    v8f zero = {0, 0, 0, 0, 0, 0, 0, 0};
#pragma unroll
    for (int h = 0; h < 4; ++h) {
      oacc[h] = zero;
#pragma unroll
      for (int r = 0; r < 8; ++r) { mrow[h][r] = -INFINITY; lrow[h][r] = 0.f; }
    }
    for (int kt = 0; kt < 21; ++kt) {
      int kbse = kt * 16;
      int nk = kbse + m16;
      int wsel = nk >> 5, bsel = nk & 31;
      float maskf[8];
#pragma unroll
      for (int r = 0; r < 8; ++r) {
        int nq = q0 + rbase + r; if (nq > NSEQ - 1) nq = NSEQ - 1;
        unsigned w = mS[nq * 12 + wsel];
        maskf[r] = ((w >> bsel) & 1u) ? 0.f : -INFINITY;  /* padded nk -> bit 0 -> -inf */
      }
#pragma unroll
      for (int h = 0; h < 4; ++h) {
        v8f s = zero;
#pragma unroll
        for (int kc = 0; kc < 4; ++kc) {
          const float* kp = &kS[(size_t)(kbse + m16) * KSTR + h * 16 + kc * 4 + koff];
          v2f b = {kp[0], kp[1]};
          s = wmma4(qa[h][kc], b, s);
        }
        float pv[8];
#pragma unroll
        for (int r = 0; r < 8; ++r) {
          float t = s[r] * 0.25f + maskf[r];           /* 1/sqrt(hd=16) */
          float mx = t;
          mx = fmaxf(mx, __shfl_xor(mx, 1, 32));
          mx = fmaxf(mx, __shfl_xor(mx, 2, 32));
          mx = fmaxf(mx, __shfl_xor(mx, 4, 32));
          mx = fmaxf(mx, __shfl_xor(mx, 8, 32));
          float nm = fmaxf(mrow[h][r], mx);
          float corr = (nm == -INFINITY) ? 1.f : __expf(mrow[h][r] - nm);
          float p = (t == -INFINITY) ? 0.f : __expf(t - nm);
          float rs = p;
          rs += __shfl_xor(rs, 1, 32);
          rs += __shfl_xor(rs, 2, 32);
          rs += __shfl_xor(rs, 4, 32);
          rs += __shfl_xor(rs, 8, 32);
          lrow[h][r] = lrow[h][r] * corr + rs;
          mrow[h][r] = nm;
          oacc[h][r] *= corr;
          pv[r] = p;
        }
        /* C-layout P tile -> LDS -> A-layout fragments */
#pragma unroll
        for (int r = 0; r < 8; ++r) pw[(rbase + r) * 17 + m16] = pv[r];
        asm volatile("s_wait_dscnt 0" ::: "memory");
#pragma unroll
        for (int kc = 0; kc < 4; ++kc) {
          v2f pa = {pw[m16 * 17 + kc * 4 + koff], pw[m16 * 17 + kc * 4 + koff + 1]};
          v2f b = {vS[(size_t)(kbse + kc * 4 + koff) * KSTR + h * 16 + m16],
                   vS[(size_t)(kbse + kc * 4 + koff + 1) * KSTR + h * 16 + m16]};
          oacc[h] = wmma4(pa, b, oacc[h]);
        }
        asm volatile("s_wait_dscnt 0" ::: "memory");
      }
    }
#pragma unroll
    for (int h = 0; h < 4; ++h)
#pragma unroll
      for (int r = 0; r < 8; ++r) {
        int nq = q0 + rbase + r;
        if (nq < NSEQ) {
          float inv = 1.f / lrow[h][r];
          ob[((size_t)bp * NSEQ + nq) * 64 + h * 16 + m16] = oacc[h][r] * inv;
        }
      }
  }
}

/* ------------------------------------------------------------------ */
/* Kernel 5: y = gelu(o @ W2 + b2), scatter to (B,N,P,C)               */
/* ------------------------------------------------------------------ */
__global__ void outproj_kernel(const float* __restrict__ ob,
                               const float* __restrict__ ws,
                               float* __restrict__ out) {
  __shared__ float sW[4160];
  for (int i = threadIdx.x; i < 4160; i += 256) sW[i] = ws[W2_OFF + i];
  __syncthreads();
  int wave = threadIdx.x >> 5, lane = threadIdx.x & 31;
  int m16 = lane & 15, koff = (lane >> 4) * 2, rbase = (lane >> 4) * 8;
  long row0 = ((long)blockIdx.x * 8 + wave) * 16;
  v2f a[16];
  const float* orow = ob + (row0 + m16) * 64;
#pragma unroll
  for (int kc = 0; kc < 16; ++kc) a[kc] = *(const v2f*)(orow + kc * 4 + koff);
#pragma unroll
  for (int nt = 0; nt < 4; ++nt) {
    int n = nt * 16 + m16;
    float bv = sW[4096 + n];
    v8f c = {bv, bv, bv, bv, bv, bv, bv, bv};
#pragma unroll
    for (int kc = 0; kc < 16; ++kc) {
      v2f b = {sW[(kc * 4 + koff) * 64 + n], sW[(kc * 4 + koff + 1) * 64 + n]};
      c = wmma4(a[kc], b, c);
    }
#pragma unroll
    for (int r = 0; r < 8; ++r) {
      int R = (int)row0 + rbase + r;
      int bpv = R / NSEQ;
      int nn = R - bpv * NSEQ;
      int bb = bpv >> 6, pp = bpv & 63;
      float vv = c[r];
      float g = 0.5f * vv * (1.f + erff(vv * 0.70710678118654752f)); /* exact gelu */
      out[(((size_t)bb * NSEQ + nn) * 64 + pp) * 64 + n] = g;
    }
  }
}

/* ------------------------------------------------------------------ */
extern "C" void kernel_launch(void* const* d_in, const int* in_sizes, int n_in,
                              void* d_out, int out_size, void* d_ws, size_t ws_size,
                              hipStream_t stream) {
  (void)in_sizes; (void)n_in; (void)out_size; (void)ws_size;
  const float* x      = (const float*)d_in[0];
  const int*   adj    = (const int*)d_in[1];
  const float* W_in   = (const float*)d_in[2];
  const float* b_in   = (const float*)d_in[3];
  const float* ipw    = (const float*)d_in[4];
  const float* ipb    = (const float*)d_in[5];
  const float* out_w  = (const float*)d_in[6];
  const float* out_b  = (const float*)d_in[7];
  const float* W_out  = (const float*)d_in[8];
  const float* b_out  = (const float*)d_in[9];
  float* ws  = (float*)d_ws;
  float* out = (float*)d_out;

  float* qbuf = ws + QBUF_OFF;
  float* kbuf = ws + KBUF_OFF;
  float* vbuf = ws + VBUF_OFF;
  unsigned* mbuf = (unsigned*)(ws + MASK_OFF);

  prep_weights<<<1, 256, 0, stream>>>(W_in, b_in, ipw, ipb, out_w, out_b, W_out,
                                      b_out, ws);
  prep_mask<<<14, 256, 0, stream>>>(adj, mbuf);
  qkv_kernel<<<1300, 256, 0, stream>>>(x, ws, qbuf, kbuf, vbuf);

  size_t attn_lds = (size_t)(2 * KROWS * KSTR + NSEQ * 12 + 8 * 272) * 4;
  (void)hipFuncSetAttribute((const void*)attn_kernel,
                            hipFuncAttributeMaxDynamicSharedMemorySize,
                            (int)attn_lds);
  /* o aliases qbuf: each wave consumes its q tile before writing its o tile */
  attn_kernel<<<512, 256, attn_lds, stream>>>(qbuf, kbuf, vbuf, mbuf, qbuf);

  outproj_kernel<<<1300, 256, 0, stream>>>(qbuf, ws, out);
}